// GCNModel_80625126080586
// MI455X (gfx1250) — compile-verified
//
#include <hip/hip_runtime.h>

#define N_NODES 100000
#define N_EDGES 1600000
#define F_IN    512
#define HIDDEN  16
#define N_CLS   7

typedef float v2f __attribute__((ext_vector_type(2)));
typedef float v8f __attribute__((ext_vector_type(8)));

// ---------------- degree / dinv ----------------
__global__ void k_deg_init(float* deg, int n) {
    int t = blockIdx.x * blockDim.x + threadIdx.x;
    if (t < n) deg[t] = 1.0f;               // self-loop contributes 1
}

__global__ void k_deg_edges(const int* __restrict__ dst, float* deg, int nE) {
    int e = blockIdx.x * blockDim.x + threadIdx.x;
    if (e < nE) atomicAdd(&deg[dst[e]], 1.0f);
}

__global__ void k_dinv(float* deg, int n) {
    int t = blockIdx.x * blockDim.x + threadIdx.x;
    if (t < n) deg[t] = __frsqrt_rn(deg[t]);   // deg >= 1 always
}

// ---------------- GEMM0: h0 = X(100000x512) @ W0(512x16), fp32 WMMA ----------------
__global__ __launch_bounds__(256) void k_gemm0(const float* __restrict__ x,
                                               const float* __restrict__ W0,
                                               float* __restrict__ h0, int nTiles) {
    __shared__ float sW[F_IN * HIDDEN];      // 512x16 = 32KB
    for (int i = threadIdx.x; i < (F_IN * HIDDEN) / 4; i += 256)
        ((float4*)sW)[i] = ((const float4*)W0)[i];
    __syncthreads();

    const int wave   = threadIdx.x >> 5;
    const int lane   = threadIdx.x & 31;
    const int tile   = blockIdx.x * 8 + wave;
    if (tile >= nTiles) return;              // wave-uniform: EXEC stays all-1s

    const int laneHi = lane >> 4;            // 0 or 1
    const int lane15 = lane & 15;
    const int row    = tile * 16 + lane15;
    const float* xrow = x + (long)row * F_IN;
    const int koff   = laneHi ? 4 : 0;       // lanes 0-15: k+0..3, lanes 16-31: k+4..7

    v8f acc = {};
    for (int kb = 0; kb < F_IN; kb += 8) {
        float4 a4 = *(const float4*)(xrow + kb + koff);
        const float* wb = sW + (kb + koff) * HIDDEN + lane15;
        // WMMA #0: kslots = {kb+0, kb+1, kb+4, kb+5}
        v2f a0; a0.x = a4.x; a0.y = a4.y;
        v2f b0; b0.x = wb[0];  b0.y = wb[16];
        acc = __builtin_amdgcn_wmma_f32_16x16x4_f32(false, a0, false, b0,
                                                    (short)0, acc, false, false);
        // WMMA #1: kslots = {kb+2, kb+3, kb+6, kb+7}
        v2f a1; a1.x = a4.z; a1.y = a4.w;
        v2f b1; b1.x = wb[32]; b1.y = wb[48];
        acc = __builtin_amdgcn_wmma_f32_16x16x4_f32(false, a1, false, b1,
                                                    (short)0, acc, false, false);
    }
    // C/D layout: VGPR i -> M = (laneHi ? 8+i : i), N = lane15
    float* out = h0 + (long)tile * 256 + laneHi * 128 + lane15;
#pragma unroll
    for (int i = 0; i < 8; ++i) out[i * 16] = acc[i];
}

// ---------------- aggregation layer 0 (16 features) ----------------
__global__ void k_self_init16(const float* __restrict__ h, const float* __restrict__ dinv,
                              float* __restrict__ agg, int n4) {
    int t = blockIdx.x * blockDim.x + threadIdx.x;   // (node, quad)
    if (t >= n4) return;
    int node = t >> 2, q = t & 3;
    float w = dinv[node]; w *= w;
    float4 v = ((const float4*)(h + (long)node * 16))[q];
    v.x *= w; v.y *= w; v.z *= w; v.w *= w;
    ((float4*)(agg + (long)node * 16))[q] = v;
}

__global__ void k_edge16(const int* __restrict__ src, const int* __restrict__ dst,
                         const float* __restrict__ dinv, const float* __restrict__ h,
                         float* __restrict__ agg, int nE) {
    int e = blockIdx.x * blockDim.x + threadIdx.x;
    if (e >= nE) return;
    int s = src[e], d = dst[e];
    float w = dinv[s] * dinv[d];
    const float* hs = h + (long)s * 16;
    float* ad = agg + (long)d * 16;
#pragma unroll
    for (int i = 0; i < 16; ++i) atomicAdd(ad + i, hs[i] * w);
}

__global__ void k_relu_bias(const float* __restrict__ agg, const float* __restrict__ b0,
                            float* __restrict__ h1, int total) {
    int t = blockIdx.x * blockDim.x + threadIdx.x;
    if (t >= total) return;
    float v = agg[t] + b0[t & 15];
    h1[t] = v > 0.0f ? v : 0.0f;
}

// ---------------- GEMM1: h2 = H1(100000x16) @ W1pad(16x16), fp32 WMMA ----------------
__global__ __launch_bounds__(256) void k_gemm1(const float* __restrict__ h1,
                                               const float* __restrict__ W1,
                                               float* __restrict__ h2, int nTiles) {
    __shared__ float sW[16 * 16];            // W1 padded N: 7 -> 16 with zeros
    {
        int k = threadIdx.x >> 4, n = threadIdx.x & 15;
        sW[threadIdx.x] = (n < N_CLS) ? W1[k * N_CLS + n] : 0.0f;
    }
    __syncthreads();

    const int wave   = threadIdx.x >> 5;
    const int lane   = threadIdx.x & 31;
    const int tile   = blockIdx.x * 8 + wave;
    if (tile >= nTiles) return;

    const int laneHi = lane >> 4;
    const int lane15 = lane & 15;
    const int row    = tile * 16 + lane15;
    const float* hrow = h1 + (long)row * 16;
    const int koff   = laneHi ? 2 : 0;       // kslot2/3 come from lanes 16-31

    v8f acc = {};
#pragma unroll
    for (int kb = 0; kb < 16; kb += 4) {
        v2f a; a.x = hrow[kb + koff]; a.y = hrow[kb + koff + 1];
        v2f b; b.x = sW[(kb + koff) * 16 + lane15];
               b.y = sW[(kb + koff + 1) * 16 + lane15];
        acc = __builtin_amdgcn_wmma_f32_16x16x4_f32(false, a, false, b,
                                                    (short)0, acc, false, false);
    }
    float* out = h2 + (long)tile * 256 + laneHi * 128 + lane15;
#pragma unroll
    for (int i = 0; i < 8; ++i) out[i * 16] = acc[i];
}

// ---------------- aggregation layer 1 (7 classes) -> d_out ----------------
__global__ void k_out_init7(const float* __restrict__ h2, const float* __restrict__ dinv,
                            const float* __restrict__ b1, float* __restrict__ out, int total) {
    int t = blockIdx.x * blockDim.x + threadIdx.x;   // (node, c), total = N*7
    if (t >= total) return;
    int node = t / N_CLS, c = t - node * N_CLS;
    float w = dinv[node];
    out[t] = h2[(long)node * 16 + c] * w * w + b1[c];
}

__global__ void k_edge7(const int* __restrict__ src, const int* __restrict__ dst,
                        const float* __restrict__ dinv, const float* __restrict__ h2,
                        float* __restrict__ out, int nE) {
    int e = blockIdx.x * blockDim.x + threadIdx.x;
    if (e >= nE) return;
    int s = src[e], d = dst[e];
    float w = dinv[s] * dinv[d];
    const float* hs = h2 + (long)s * 16;
    float* od = out + (long)d * N_CLS;
#pragma unroll
    for (int c = 0; c < N_CLS; ++c) atomicAdd(od + c, hs[c] * w);
}

// ---------------- launcher ----------------
extern "C" void kernel_launch(void* const* d_in, const int* in_sizes, int n_in,
                              void* d_out, int out_size, void* d_ws, size_t ws_size,
                              hipStream_t stream) {
    const float* x  = (const float*)d_in[0];
    const int*   ei = (const int*)d_in[1];
    const float* W0 = (const float*)d_in[2];
    const float* b0 = (const float*)d_in[3];
    const float* W1 = (const float*)d_in[4];
    const float* b1 = (const float*)d_in[5];
    float* out = (float*)d_out;

    const int* src = ei;
    const int* dst = ei + N_EDGES;

    float* ws   = (float*)d_ws;
    float* dinv = ws;                         // N_NODES
    float* h0   = dinv + N_NODES;             // N_NODES*16 (reused for h1 after agg0)
    float* agg0 = h0 + (long)N_NODES * 16;    // N_NODES*16
    float* h2   = agg0 + (long)N_NODES * 16;  // N_NODES*16 (N padded 7->16)

    const int nTiles = N_NODES / 16;          // 6250 exactly
    const dim3 B(256);
    auto gr = [](long n) { return dim3((unsigned)((n + 255) / 256)); };

    // degrees -> dinv (in place)
    k_deg_init <<<gr(N_NODES), B, 0, stream>>>(dinv, N_NODES);
    k_deg_edges<<<gr(N_EDGES), B, 0, stream>>>(dst, dinv, N_EDGES);
    k_dinv     <<<gr(N_NODES), B, 0, stream>>>(dinv, N_NODES);

    // layer 0
    k_gemm0<<<dim3((nTiles + 7) / 8), B, 0, stream>>>(x, W0, h0, nTiles);
    k_self_init16<<<gr((long)N_NODES * 4), B, 0, stream>>>(h0, dinv, agg0, N_NODES * 4);
    k_edge16<<<gr(N_EDGES), B, 0, stream>>>(src, dst, dinv, h0, agg0, N_EDGES);
    k_relu_bias<<<gr((long)N_NODES * 16), B, 0, stream>>>(agg0, b0, h0, N_NODES * 16); // h1 -> h0 buf

    // layer 1
    k_gemm1<<<dim3((nTiles + 7) / 8), B, 0, stream>>>(h0, W1, h2, nTiles);
    k_out_init7<<<gr((long)N_NODES * N_CLS), B, 0, stream>>>(h2, dinv, b1, out, N_NODES * N_CLS);
    k_edge7<<<gr(N_EDGES), B, 0, stream>>>(src, dst, dinv, h2, out, N_EDGES);
}